// FrequencyAttention_6107443495343
// MI455X (gfx1250) — compile-verified
//
#include <hip/hip_runtime.h>
#include <math.h>

// FrequencyAttention for MI455X (gfx1250), fp32 end-to-end via V_WMMA_F32_16X16X4_F32.
//
// Pipeline:
//   1) BN-in batch stats  -> per-channel scale s, shift t
//   2) Fold BN into weights: W' = W*diag(s), bias = W*t
//   3) per batch b:
//        proj  : q/k/v = W' x + bias          (WMMA GEMM, K=512, LDS-tiled)
//        fft   : F X F per 64x64 image        (WMMA, DFT-matrix method)
//        attn  : complex QK^T (K=4096) + temperature + re/im softmax
//                 -> async-to-LDS double-buffered staging (ASYNCcnt)
//        out   : attn @ V (complex)
//        ifft  : (1/4096) Re[conj(F) Z conj(F)] ; out_pre = x + gamma*...
//   4) BN-out stats + apply -> d_out
//
// Fragment layouts per CDNA5 ISA 7.12.2:
//   A(16x4): lane%16 = M row, VGPR j holds K = k0 + (lane/16)*2 + j.
//   C/D:     lane%16 = N,     VGPR j holds M = m0 + (lane/16)*8 + j.
//   B(4x16)  assumed mirror of A: lane%16 = N, VGPR j: K = k0+(lane/16)*2+j.

#define HW   4096
#define CCH  512
#define NB   8
#define NHD  8
#define EPSW 1e-5f

typedef __attribute__((ext_vector_type(2))) float v2f;
typedef __attribute__((ext_vector_type(8))) float v8f;

__device__ __forceinline__ v8f wmma_f32(v2f a, v2f b, v8f c) {
  return __builtin_amdgcn_wmma_f32_16x16x4_f32(false, a, false, b, (short)0, c, false, false);
}

// A fragment: M rows = m0..m0+15 (lane%16), K = k0 + (lane/16)*2 + j
__device__ __forceinline__ v2f frag_a(const float* M, int ld, int m0, int k0, int lane) {
  int row = m0 + (lane & 15);
  int kk  = k0 + ((lane >> 4) << 1);
  v2f r;
  r.x = M[(size_t)row * ld + kk];
  r.y = M[(size_t)row * ld + kk + 1];
  return r;
}

// B fragment: N cols = n0..n0+15 (lane%16), K = k0 + (lane/16)*2 + j  (B stored [K][N])
__device__ __forceinline__ v2f frag_b(const float* M, int ld, int k0, int n0, int lane) {
  int n  = n0 + (lane & 15);
  int kk = k0 + ((lane >> 4) << 1);
  v2f r;
  r.x = M[(size_t)kk * ld + n];
  r.y = M[(size_t)(kk + 1) * ld + n];
  return r;
}

// C/D store: D[m][n], row-major with leading dim ld
__device__ __forceinline__ void stC(float* D, int ld, int m0, int n0, int lane, v8f c) {
  int n  = n0 + (lane & 15);
  int mb = m0 + ((lane >> 4) << 3);
#pragma unroll
  for (int j = 0; j < 8; ++j) D[(size_t)(mb + j) * ld + n] = c[j];
}

// ---------------------------------------------------------------- DFT matrix
__global__ void k_dft_init(float* __restrict__ Fre, float* __restrict__ Fim) {
  int t = blockIdx.x * 256 + threadIdx.x;
  if (t < 64 * 64) {
    int j = t >> 6, k = t & 63;
    double ang = -2.0 * 3.14159265358979323846 * (double)((j * k) & 63) / 64.0;
    Fre[t] = (float)cos(ang);
    Fim[t] = (float)sin(ang);
  }
}

// ------------------------------------------- BN batch stats -> scale/shift
__global__ void k_bn_stats(const float* __restrict__ x, const float* __restrict__ w,
                           const float* __restrict__ b, float* __restrict__ s,
                           float* __restrict__ t) {
  int c = blockIdx.x, tid = threadIdx.x;
  float sum = 0.f, sq = 0.f;
  for (int bb = 0; bb < NB; ++bb) {
    const float* p = x + ((size_t)bb * CCH + c) * HW;
    for (int i = tid; i < HW; i += 256) { float v = p[i]; sum += v; sq += v * v; }
  }
  __shared__ float r1[256], r2[256];
  r1[tid] = sum; r2[tid] = sq;
  __syncthreads();
  for (int st = 128; st > 0; st >>= 1) {
    if (tid < st) { r1[tid] += r1[tid + st]; r2[tid] += r2[tid + st]; }
    __syncthreads();
  }
  if (tid == 0) {
    const float invn = 1.0f / (float)(NB * HW);
    float mean = r1[0] * invn;
    float var  = r2[0] * invn - mean * mean;
    float sc   = w[c] * rsqrtf(var + EPSW);
    s[c] = sc;
    t[c] = b[c] - mean * sc;
  }
}

// --------------------------------- fold BN-in into the three conv weights
__global__ void k_fold_w(const float* __restrict__ Wq, const float* __restrict__ Wk,
                         const float* __restrict__ Wv, const float* __restrict__ s,
                         const float* __restrict__ t, float* __restrict__ W2,
                         float* __restrict__ biases) {
  int o = blockIdx.x, z = blockIdx.y, tid = threadIdx.x;
  const float* W = (z == 0) ? Wq : (z == 1) ? Wk : Wv;
  float* Wd = W2 + (size_t)z * CCH * CCH;
  float acc = 0.f;
  for (int c = tid; c < CCH; c += 256) {
    float w = W[(size_t)o * CCH + c];
    Wd[(size_t)o * CCH + c] = w * s[c];
    acc += w * t[c];
  }
  __shared__ float r[256];
  r[tid] = acc;
  __syncthreads();
  for (int st = 128; st > 0; st >>= 1) {
    if (tid < st) r[tid] += r[tid + st];
    __syncthreads();
  }
  if (tid == 0) biases[(size_t)z * CCH + o] = r[0];
}

// ------------------------------------- projections: spat[z] = W2[z] * x[b] + bias
__global__ __launch_bounds__(256)
void k_proj(const float* __restrict__ x, const float* __restrict__ W2,
            const float* __restrict__ biases, float* __restrict__ spat, int b) {
  const int nt = blockIdx.x, ot = blockIdx.y, z = blockIdx.z;
  const int tid = threadIdx.x, lane = tid & 31, wave = tid >> 5;
  const float* W  = W2 + (size_t)z * CCH * CCH;
  const float* bi = biases + (size_t)z * CCH;
  const float* xb = x + (size_t)b * CCH * HW;
  float* out = spat + (size_t)z * CCH * HW;
  const int o0 = ot * 64, n0g = nt * 64;

  __shared__ float As[64 * 33];  // W tile  [m][k]
  __shared__ float Bs[32 * 65];  // x tile  [k][n]

  const int m0  = (wave & 3) * 16;
  const int n0a = ((wave >> 2) * 2) * 16;
  const int n0b = n0a + 16;

  v8f acc0 = {}, acc1 = {};
  for (int kb = 0; kb < CCH; kb += 32) {
    __syncthreads();
    for (int i = tid; i < 64 * 32; i += 256) {
      int m = i >> 5, kk = i & 31;
      As[m * 33 + kk] = W[(size_t)(o0 + m) * CCH + kb + kk];
    }
    for (int i = tid; i < 32 * 64; i += 256) {
      int kk = i >> 6, n = i & 63;
      Bs[kk * 65 + n] = xb[(size_t)(kb + kk) * HW + n0g + n];
    }
    __syncthreads();
#pragma unroll
    for (int kk = 0; kk < 32; kk += 4) {
      v2f a  = frag_a(As, 33, m0, kk, lane);
      v2f b0 = frag_b(Bs, 65, kk, n0a, lane);
      v2f b1 = frag_b(Bs, 65, kk, n0b, lane);
      acc0 = wmma_f32(a, b0, acc0);
      acc1 = wmma_f32(a, b1, acc1);
    }
  }
  int n  = lane & 15;
  int mb = (lane >> 4) << 3;
#pragma unroll
  for (int j = 0; j < 8; ++j) {
    int m = m0 + mb + j;
    float bv = bi[o0 + m];
    out[(size_t)(o0 + m) * HW + n0g + n0a + n] = acc0[j] + bv;
    out[(size_t)(o0 + m) * HW + n0g + n0b + n] = acc1[j] + bv;
  }
}

// --------------------------------------- fft2 via DFT matmul:  Y = F * X * F
__global__ __launch_bounds__(256)
void k_fft(const float* __restrict__ spat, const float* __restrict__ Fre,
           const float* __restrict__ Fim, float* __restrict__ cplx) {
  const int c = blockIdx.x, z = blockIdx.y;
  const int tid = threadIdx.x, lane = tid & 31, wave = tid >> 5;
  const size_t CHW = (size_t)CCH * HW;
  const float* X = spat + ((size_t)z * CCH + c) * HW;  // 64x64 image
  float* Yre = cplx + ((size_t)(z * 2)) * CHW + (size_t)c * HW;
  float* Yim = cplx + ((size_t)(z * 2 + 1)) * CHW + (size_t)c * HW;

  __shared__ float Tre[64 * 65];
  __shared__ float Tim[64 * 65];

  const int m0  = (wave & 3) * 16;
  const int n0a = ((wave >> 2) * 2) * 16;
  const int n0b = n0a + 16;

  // stage 1: T = X * F   (X real)
  v8f tr0 = {}, tr1 = {}, ti0 = {}, ti1 = {};
#pragma unroll
  for (int kk = 0; kk < 64; kk += 4) {
    v2f a   = frag_a(X, 64, m0, kk, lane);
    v2f br0 = frag_b(Fre, 64, kk, n0a, lane);
    v2f br1 = frag_b(Fre, 64, kk, n0b, lane);
    v2f bi0 = frag_b(Fim, 64, kk, n0a, lane);
    v2f bi1 = frag_b(Fim, 64, kk, n0b, lane);
    tr0 = wmma_f32(a, br0, tr0); tr1 = wmma_f32(a, br1, tr1);
    ti0 = wmma_f32(a, bi0, ti0); ti1 = wmma_f32(a, bi1, ti1);
  }
  stC(Tre, 65, m0, n0a, lane, tr0); stC(Tre, 65, m0, n0b, lane, tr1);
  stC(Tim, 65, m0, n0a, lane, ti0); stC(Tim, 65, m0, n0b, lane, ti1);
  __syncthreads();

  // stage 2: Y = F * T,  Yre = Fre*Tre - Fim*Tim ; Yim = Fre*Tim + Fim*Tre
  v8f prr0 = {}, prr1 = {}, pii0 = {}, pii1 = {}, yim0 = {}, yim1 = {};
#pragma unroll
  for (int kk = 0; kk < 64; kk += 4) {
    v2f are  = frag_a(Fre, 64, m0, kk, lane);
    v2f aim  = frag_a(Fim, 64, m0, kk, lane);
    v2f btr0 = frag_b(Tre, 65, kk, n0a, lane);
    v2f btr1 = frag_b(Tre, 65, kk, n0b, lane);
    v2f bti0 = frag_b(Tim, 65, kk, n0a, lane);
    v2f bti1 = frag_b(Tim, 65, kk, n0b, lane);
    prr0 = wmma_f32(are, btr0, prr0); prr1 = wmma_f32(are, btr1, prr1);
    pii0 = wmma_f32(aim, bti0, pii0); pii1 = wmma_f32(aim, bti1, pii1);
    yim0 = wmma_f32(are, bti0, yim0); yim0 = wmma_f32(aim, btr0, yim0);
    yim1 = wmma_f32(are, bti1, yim1); yim1 = wmma_f32(aim, btr1, yim1);
  }
  v8f yre0, yre1;
#pragma unroll
  for (int j = 0; j < 8; ++j) { yre0[j] = prr0[j] - pii0[j]; yre1[j] = prr1[j] - pii1[j]; }
  stC(Yre, 64, m0, n0a, lane, yre0); stC(Yre, 64, m0, n0b, lane, yre1);
  stC(Yim, 64, m0, n0a, lane, yim0); stC(Yim, 64, m0, n0b, lane, yim1);
}

// ------------- attention scores: complex Q K^T over n=4096, + softmax(re/im)
// Double-buffered async-to-LDS staging: GLOBAL_LOAD_ASYNC_TO_LDS_B128 (ASYNCcnt).
__global__ __launch_bounds__(256)
void k_attn(const float* __restrict__ cplx, const float* __restrict__ temperature,
            float* __restrict__ attnG) {
  const int h = blockIdx.x;
  const int tid = threadIdx.x, lane = tid & 31, wave = tid >> 5;
  const size_t CHW = (size_t)CCH * HW;
  const float* gb0 = cplx + 0 * CHW + (size_t)h * 64 * HW;  // Qre
  const float* gb1 = cplx + 1 * CHW + (size_t)h * 64 * HW;  // Qim
  const float* gb2 = cplx + 2 * CHW + (size_t)h * 64 * HW;  // Kre
  const float* gb3 = cplx + 3 * CHW + (size_t)h * 64 * HW;  // Kim

  // 2 ping-pong buffers x 4 matrices x [64][33]
  __shared__ float smem[2 * 4 * 2112];

  const float* gb[4] = {gb0, gb1, gb2, gb3};

  // Stage chunk [kb, kb+32) of all 4 matrices into buffer p.
  // 2048 float4 transfers; thread covers 8: per unrolled step, matrix = it>>1
  // is compile-time (tid in [0,256) never crosses a 512-float4 boundary).
  auto stage = [&](int p, int kb) {
#pragma unroll
    for (int it = 0; it < 8; ++it) {
      const int i = tid + it * 256;
      const int m = it >> 1;                 // matrix index (constant per step)
      const int r = (i & 511) >> 3;          // row 0..63
      const int kq = (i & 7) << 2;           // k offset 0,4,...,28
      const float* g = gb[m] + (size_t)r * HW + kb + kq;
      float* l = smem + p * 8448 + m * 2112 + r * 33 + kq;
      unsigned loff = (unsigned)(unsigned long long)l;  // flat->LDS: addr[31:0]
      asm volatile("global_load_async_to_lds_b128 %0, %1, off"
                   :: "v"(loff), "v"(g)
                   : "memory");
    }
  };

  const int m0  = (wave & 3) * 16;
  const int n0a = ((wave >> 2) * 2) * 16;
  const int n0b = n0a + 16;

  v8f prr0 = {}, prr1 = {}, pii0 = {}, pii1 = {}, aim0 = {}, aim1 = {};

  stage(0, 0);
  for (int kb = 0; kb < HW; kb += 32) {
    const int p = (kb >> 5) & 1;
    // my async writes for buffer p are done; barrier makes everyone's visible
    asm volatile("s_wait_asynccnt 0x0" ::: "memory");
    __syncthreads();
    if (kb + 32 < HW) stage(p ^ 1, kb + 32);  // prefetch next chunk

    const float* Qre_s = smem + p * 8448;
    const float* Qim_s = Qre_s + 2112;
    const float* Kre_s = Qre_s + 4224;
    const float* Kim_s = Qre_s + 6336;
#pragma unroll
    for (int kk = 0; kk < 32; kk += 4) {
      v2f are = frag_a(Qre_s, 33, m0, kk, lane);
      v2f aqi = frag_a(Qim_s, 33, m0, kk, lane);
      // K stored [d][n]; B^T load == frag_a with row = output column d
      v2f br0 = frag_a(Kre_s, 33, n0a, kk, lane);
      v2f br1 = frag_a(Kre_s, 33, n0b, kk, lane);
      v2f bi0 = frag_a(Kim_s, 33, n0a, kk, lane);
      v2f bi1 = frag_a(Kim_s, 33, n0b, kk, lane);
      prr0 = wmma_f32(are, br0, prr0); prr1 = wmma_f32(are, br1, prr1);
      pii0 = wmma_f32(aqi, bi0, pii0); pii1 = wmma_f32(aqi, bi1, pii1);
      aim0 = wmma_f32(are, bi0, aim0); aim0 = wmma_f32(aqi, br0, aim0);
      aim1 = wmma_f32(are, bi1, aim1); aim1 = wmma_f32(aqi, br1, aim1);
    }
  }
  __syncthreads();
  // combine + temperature, stash 64x64 re/im into LDS (reuse buffer 0)
  const float temp = temperature[h];
  float* Are_l = smem;          // [64][64]
  float* Aim_l = smem + 4096;   // [64][64]
  v8f re0, re1, im0, im1;
#pragma unroll
  for (int j = 0; j < 8; ++j) {
    re0[j] = (prr0[j] - pii0[j]) * temp;
    re1[j] = (prr1[j] - pii1[j]) * temp;
    im0[j] = aim0[j] * temp;
    im1[j] = aim1[j] * temp;
  }
  stC(Are_l, 64, m0, n0a, lane, re0); stC(Are_l, 64, m0, n0b, lane, re1);
  stC(Aim_l, 64, m0, n0a, lane, im0); stC(Aim_l, 64, m0, n0b, lane, im1);
  __syncthreads();
  // softmax over d (64) independently for re and im
  if (tid < 128) {
    int part = tid >> 6, row = tid & 63;
    const float* src = (part == 0 ? Are_l : Aim_l) + row * 64;
    float mx = src[0];
    for (int d = 1; d < 64; ++d) mx = fmaxf(mx, src[d]);
    float s = 0.f;
    for (int d = 0; d < 64; ++d) s += expf(src[d] - mx);
    float inv = 1.0f / s;
    float* dst = attnG + ((size_t)part * NHD + h) * 4096 + (size_t)row * 64;
    for (int d = 0; d < 64; ++d) dst[d] = expf(src[d] - mx) * inv;
  }
}

// ------------------------------------------- out_f = attn @ V (complex, K=64)
__global__ __launch_bounds__(256)
void k_out(const float* __restrict__ attnG, const float* __restrict__ cplx,
           float* __restrict__ outf) {
  const int nt = blockIdx.x, h = blockIdx.y;
  const int tid = threadIdx.x, lane = tid & 31, wave = tid >> 5;
  const size_t CHW = (size_t)CCH * HW;
  const float* Are = attnG + ((size_t)0 * NHD + h) * 4096;  // [64][64]
  const float* Aim = attnG + ((size_t)1 * NHD + h) * 4096;
  const float* Vre = cplx + 4 * CHW + (size_t)h * 64 * HW;
  const float* Vim = cplx + 5 * CHW + (size_t)h * 64 * HW;
  float* Ore = outf + (size_t)h * 64 * HW;
  float* Oim = outf + CHW + (size_t)h * 64 * HW;

  const int n0g = nt * 64;
  const int m0  = (wave & 3) * 16;
  const int n0a = n0g + ((wave >> 2) * 2) * 16;
  const int n0b = n0a + 16;

  v8f prr0 = {}, prr1 = {}, pii0 = {}, pii1 = {}, pim0 = {}, pim1 = {};
#pragma unroll
  for (int kk = 0; kk < 64; kk += 4) {
    v2f are = frag_a(Are, 64, m0, kk, lane);
    v2f aim = frag_a(Aim, 64, m0, kk, lane);
    v2f br0 = frag_b(Vre, HW, kk, n0a, lane);
    v2f br1 = frag_b(Vre, HW, kk, n0b, lane);
    v2f bi0 = frag_b(Vim, HW, kk, n0a, lane);
    v2f bi1 = frag_b(Vim, HW, kk, n0b, lane);
    prr0 = wmma_f32(are, br0, prr0); prr1 = wmma_f32(are, br1, prr1);
    pii0 = wmma_f32(aim, bi0, pii0); pii1 = wmma_f32(aim, bi1, pii1);
    pim0 = wmma_f32(are, bi0, pim0); pim0 = wmma_f32(aim, br0, pim0);
    pim1 = wmma_f32(are, bi1, pim1); pim1 = wmma_f32(aim, br1, pim1);
  }
  v8f re0, re1;
#pragma unroll
  for (int j = 0; j < 8; ++j) { re0[j] = prr0[j] - pii0[j]; re1[j] = prr1[j] - pii1[j]; }
  stC(Ore, HW, m0, n0a, lane, re0); stC(Ore, HW, m0, n0b, lane, re1);
  stC(Oim, HW, m0, n0a, lane, pim0); stC(Oim, HW, m0, n0b, lane, pim1);
}

// --- ifft2 (real part) via conj-DFT matmul + residual: out_pre = x + g*ifft
__global__ __launch_bounds__(256)
void k_ifft(const float* __restrict__ outf, const float* __restrict__ Fre,
            const float* __restrict__ Fim, const float* __restrict__ x,
            const float* __restrict__ gamma, float* __restrict__ out_pre, int b) {
  const int c = blockIdx.x;
  const int tid = threadIdx.x, lane = tid & 31, wave = tid >> 5;
  const size_t CHW = (size_t)CCH * HW;
  const float* Zre = outf + (size_t)c * HW;
  const float* Zim = outf + CHW + (size_t)c * HW;
  const float* xb  = x + ((size_t)b * CCH + c) * HW;
  float* op = out_pre + ((size_t)b * CCH + c) * HW;

  __shared__ float Ure[64 * 65];
  __shared__ float Uim[64 * 65];

  const int m0  = (wave & 3) * 16;
  const int n0a = ((wave >> 2) * 2) * 16;
  const int n0b = n0a + 16;

  // stage 1: U = Z * conj(F): Ure = Zre*Fre + Zim*Fim ; Uim = Zim*Fre - Zre*Fim
  v8f ur0 = {}, ur1 = {}, p10 = {}, p11 = {}, p20 = {}, p21 = {};
#pragma unroll
  for (int kk = 0; kk < 64; kk += 4) {
    v2f azr = frag_a(Zre, 64, m0, kk, lane);
    v2f azi = frag_a(Zim, 64, m0, kk, lane);
    v2f br0 = frag_b(Fre, 64, kk, n0a, lane);
    v2f br1 = frag_b(Fre, 64, kk, n0b, lane);
    v2f bi0 = frag_b(Fim, 64, kk, n0a, lane);
    v2f bi1 = frag_b(Fim, 64, kk, n0b, lane);
    ur0 = wmma_f32(azr, br0, ur0); ur0 = wmma_f32(azi, bi0, ur0);
    ur1 = wmma_f32(azr, br1, ur1); ur1 = wmma_f32(azi, bi1, ur1);
    p10 = wmma_f32(azi, br0, p10); p11 = wmma_f32(azi, br1, p11);
    p20 = wmma_f32(azr, bi0, p20); p21 = wmma_f32(azr, bi1, p21);
  }
  v8f ui0, ui1;
#pragma unroll
  for (int j = 0; j < 8; ++j) { ui0[j] = p10[j] - p20[j]; ui1[j] = p11[j] - p21[j]; }
  stC(Ure, 65, m0, n0a, lane, ur0); stC(Ure, 65, m0, n0b, lane, ur1);
  stC(Uim, 65, m0, n0a, lane, ui0); stC(Uim, 65, m0, n0b, lane, ui1);
  __syncthreads();

  // stage 2 (real part only): Vre = Fre*Ure + Fim*Uim
  v8f vr0 = {}, vr1 = {};
#pragma unroll
  for (int kk = 0; kk < 64; kk += 4) {
    v2f are = frag_a(Fre, 64, m0, kk, lane);
    v2f aim = frag_a(Fim, 64, m0, kk, lane);
    v2f b0r = frag_b(Ure, 65, kk, n0a, lane);
    v2f b1r = frag_b(Ure, 65, kk, n0b, lane);
    v2f b0i = frag_b(Uim, 65, kk, n0a, lane);
    v2f b1i = frag_b(Uim, 65, kk, n0b, lane);
    vr0 = wmma_f32(are, b0r, vr0); vr0 = wmma_f32(aim, b0i, vr0);
    vr1 = wmma_f32(are, b1r, vr1); vr1 = wmma_f32(aim, b1i, vr1);
  }
  const float g = gamma[0];
  const float inv = 1.0f / 4096.0f;
  int n  = lane & 15;
  int mb = (lane >> 4) << 3;
#pragma unroll
  for (int j = 0; j < 8; ++j) {
    int m = m0 + mb + j;
    int i0 = m * 64 + n0a + n;
    int i1 = m * 64 + n0b + n;
    op[i0] = xb[i0] + g * (vr0[j] * inv);
    op[i1] = xb[i1] + g * (vr1[j] * inv);
  }
}

// ------------------------------------------------------- final BN-out apply
__global__ void k_bn_apply(const float* __restrict__ xin, const float* __restrict__ s,
                           const float* __restrict__ t, float* __restrict__ out) {
  size_t i = (size_t)blockIdx.x * 256 + threadIdx.x;
  int c = (int)((i >> 12) & 511);
  out[i] = xin[i] * s[c] + t[c];
}

extern "C" void kernel_launch(void* const* d_in, const int* in_sizes, int n_in,
                              void* d_out, int out_size, void* d_ws, size_t ws_size,
                              hipStream_t stream) {
  const float* x        = (const float*)d_in[0];
  const float* q_w      = (const float*)d_in[1];
  const float* k_w      = (const float*)d_in[2];
  const float* v_w      = (const float*)d_in[3];
  const float* temp     = (const float*)d_in[4];
  const float* gamma    = (const float*)d_in[5];
  const float* bn_in_w  = (const float*)d_in[6];
  const float* bn_in_b  = (const float*)d_in[7];
  const float* bn_out_w = (const float*)d_in[8];
  const float* bn_out_b = (const float*)d_in[9];

  float* ws = (float*)d_ws;
  const size_t CHW = (size_t)CCH * HW;
  float* Fre     = ws;                       // 4096
  float* Fim     = Fre + 4096;               // 4096
  float* s_in    = Fim + 4096;               // 512
  float* t_in    = s_in + 512;               // 512
  float* W2      = t_in + 512;               // 3*512*512
  float* biases  = W2 + (size_t)3 * CCH * CCH;   // 3*512
  float* spat    = biases + 3 * CCH;         // 3*CHW
  float* cplx    = spat + 3 * CHW;           // 6*CHW (Qre,Qim,Kre,Kim,Vre,Vim)
  float* attnG   = cplx + 6 * CHW;           // 2*8*4096
  float* outf    = attnG + (size_t)2 * NHD * 4096;  // 2*CHW
  float* out_pre = outf + 2 * CHW;           // 8*CHW
  float* s_out   = out_pre + (size_t)NB * CHW;
  float* t_out   = s_out + 512;
  (void)ws_size; (void)in_sizes; (void)n_in; (void)out_size;

  k_dft_init<<<16, 256, 0, stream>>>(Fre, Fim);
  k_bn_stats<<<512, 256, 0, stream>>>(x, bn_in_w, bn_in_b, s_in, t_in);
  k_fold_w<<<dim3(512, 3), 256, 0, stream>>>(q_w, k_w, v_w, s_in, t_in, W2, biases);

  for (int b = 0; b < NB; ++b) {
    k_proj<<<dim3(64, 8, 3), 256, 0, stream>>>(x, W2, biases, spat, b);
    k_fft<<<dim3(512, 3), 256, 0, stream>>>(spat, Fre, Fim, cplx);
    k_attn<<<NHD, 256, 0, stream>>>(cplx, temp, attnG);
    k_out<<<dim3(64, NHD), 256, 0, stream>>>(attnG, cplx, outf);
    k_ifft<<<512, 256, 0, stream>>>(outf, Fre, Fim, x, gamma, out_pre, b);
  }

  k_bn_stats<<<512, 256, 0, stream>>>(out_pre, bn_out_w, bn_out_b, s_out, t_out);
  k_bn_apply<<<65536, 256, 0, stream>>>(out_pre, s_out, t_out, (float*)d_out);
}